// uem_net_49478023250581
// MI455X (gfx1250) — compile-verified
//
#include <hip/hip_runtime.h>
#include <math.h>

typedef __attribute__((ext_vector_type(2))) float v2f;
typedef __attribute__((ext_vector_type(8))) float v8f;

#define HL 512      // xl spatial
#define HHS 256     // xh spatial
#define NB 8        // batch

__device__ __forceinline__ float gelu_exact(float x) {
    return 0.5f * x * (1.0f + erff(x * 0.7071067811865475f));
}

// ---------------------------------------------------------------------------
// Kernel 0: pre_project 1x1 conv 16 -> 8 channels at 256x256 (into workspace)
// ---------------------------------------------------------------------------
__global__ __launch_bounds__(256)
void pre_project_kernel(const float* __restrict__ xh,
                        const float* __restrict__ pre_w,
                        const float* __restrict__ pre_b,
                        float* __restrict__ xh_p) {
    int idx = blockIdx.x * 256 + threadIdx.x;       // 8*8*65536 threads
    int pos = idx & 65535;
    int o   = (idx >> 16) & 7;
    int b   = idx >> 19;
    const float* src = xh + ((size_t)b * 16) * 65536 + pos;
    float acc = pre_b[o];
#pragma unroll
    for (int c = 0; c < 16; ++c)
        acc += pre_w[o * 16 + c] * src[(size_t)c * 65536];
    xh_p[idx] = acc;
}

// ---------------------------------------------------------------------------
// Fused mega-kernel: per 16x16 tile of the 512x512 output.
//  - on-the-fly bilinear upsample of xh_p (align_corners=True)
//  - 4 branches: channel-LN -> {base dw3x3, Haar DWT -> dw3x3 -> iDWT} -> 5x5 pw
//  - tail LN(20) -> combined [tail(8x20) | res(8x9)] GEMM via fp32 WMMA 16x16x4
//  - GELU + residual add, store final output. Zero large intermediates.
// ---------------------------------------------------------------------------
__global__ __launch_bounds__(256)
void fused_kernel(const float* __restrict__ xh_p,
                  const float* __restrict__ xl,
                  const float* __restrict__ mask,
                  const float* __restrict__ g_ln_w, const float* __restrict__ g_ln_b,
                  const float* __restrict__ g_base_w, const float* __restrict__ g_base_b,
                  const float* __restrict__ g_base_s,
                  const float* __restrict__ g_wt_w, const float* __restrict__ g_wt_s,
                  const float* __restrict__ g_pw_w,
                  const float* __restrict__ tail_ln_w, const float* __restrict__ tail_ln_b,
                  const float* __restrict__ tail_w, const float* __restrict__ tail_b,
                  const float* __restrict__ res_w, const float* __restrict__ res_b,
                  float* __restrict__ out) {
    __shared__ float gin[5][20][20];     // LN'd branch input, 2-pixel halo
    __shared__ float coef[20][10][10];   // Haar coeffs (c*4+k), 1-cell halo
    __shared__ float wconv[20][8][8];    // wavelet conv result
    __shared__ float x20[20][257];       // 20-ch concat / LN'd (padded stride)
    __shared__ float xlc[12][257];       // raw xl (8ch) + mask + 3 zero channels
    __shared__ float wcomb[16][32];      // combined tail|res weights for WMMA A

    const int t  = threadIdx.x;
    const int b  = blockIdx.z;
    const int oy = blockIdx.y * 16;
    const int ox = blockIdx.x * 16;

    // Build combined weight matrix: rows 0-7 tail_w over k=0..19,
    // rows 8-15 res_w over k=20..28, zeros elsewhere.
    for (int i = t; i < 512; i += 256) {
        int m = i >> 5, k = i & 31;
        float v = 0.0f;
        if (m < 8) { if (k < 20) v = tail_w[m * 20 + k]; }
        else       { if (k >= 20 && k < 29) v = res_w[(m - 8) * 9 + (k - 20)]; }
        wcomb[m][k] = v;
    }
    // Zero the pad channels (k = 29..31 of the B operand) once.
    xlc[9][t] = 0.0f; xlc[10][t] = 0.0f; xlc[11][t] = 0.0f;

    for (int br = 0; br < 4; ++br) {
        __syncthreads();
        // ---- Phase A: gather raw gi (xh_u x2, xl x2, mask), channel-LN, stage
        for (int r = t; r < 400; r += 256) {
            int ry = r / 20, rx = r % 20;
            int Y = oy - 2 + ry, X = ox - 2 + rx;
            float v0 = 0.f, v1 = 0.f, v2 = 0.f, v3 = 0.f, v4 = 0.f;
            if (Y >= 0 && Y < HL && X >= 0 && X < HL) {
                const float SS = 255.0f / 511.0f;   // (n_in-1)/(n_out-1)
                float yc = (float)Y * SS, xc = (float)X * SS;
                int iy = (int)floorf(yc); iy = iy < 0 ? 0 : (iy > 254 ? 254 : iy);
                int ix = (int)floorf(xc); ix = ix < 0 ? 0 : (ix > 254 ? 254 : ix);
                float fy = yc - (float)iy, fx = xc - (float)ix;
                const float* p0 = xh_p + (((size_t)(b * 8 + 2 * br)) * HHS + iy) * HHS + ix;
                const float* p1 = p0 + (size_t)HHS * HHS;
                float w00 = (1.f - fy) * (1.f - fx), w01 = (1.f - fy) * fx;
                float w10 = fy * (1.f - fx),         w11 = fy * fx;
                v0 = w00 * p0[0] + w01 * p0[1] + w10 * p0[HHS] + w11 * p0[HHS + 1];
                v1 = w00 * p1[0] + w01 * p1[1] + w10 * p1[HHS] + w11 * p1[HHS + 1];
                size_t xli = (((size_t)(b * 8 + 2 * br)) * HL + Y) * HL + X;
                v2 = xl[xli];
                v3 = xl[xli + (size_t)HL * HL];
                v4 = mask[((size_t)b * HL + Y) * HL + X];
                // Stage raw center values for the fused tail/residual GEMM
                if (ry >= 2 && ry < 18 && rx >= 2 && rx < 18) {
                    int pix = (ry - 2) * 16 + (rx - 2);
                    xlc[2 * br + 0][pix] = v2;
                    xlc[2 * br + 1][pix] = v3;
                    if (br == 0) xlc[8][pix] = v4;
                }
                // LN over the 5 channels (eps 1e-6)
                float u = 0.2f * (v0 + v1 + v2 + v3 + v4);
                float d0 = v0 - u, d1 = v1 - u, d2 = v2 - u, d3 = v3 - u, d4 = v4 - u;
                float var = 0.2f * (d0 * d0 + d1 * d1 + d2 * d2 + d3 * d3 + d4 * d4);
                float inv = rsqrtf(var + 1e-6f);
                v0 = g_ln_w[br * 5 + 0] * d0 * inv + g_ln_b[br * 5 + 0];
                v1 = g_ln_w[br * 5 + 1] * d1 * inv + g_ln_b[br * 5 + 1];
                v2 = g_ln_w[br * 5 + 2] * d2 * inv + g_ln_b[br * 5 + 2];
                v3 = g_ln_w[br * 5 + 3] * d3 * inv + g_ln_b[br * 5 + 3];
                v4 = g_ln_w[br * 5 + 4] * d4 * inv + g_ln_b[br * 5 + 4];
            }
            gin[0][ry][rx] = v0; gin[1][ry][rx] = v1; gin[2][ry][rx] = v2;
            gin[3][ry][rx] = v3; gin[4][ry][rx] = v4;
        }
        __syncthreads();
        // ---- Phase B: Haar DWT coefficients (OOB halo is zero via gin zeros)
        for (int r = t; r < 500; r += 256) {
            int c = r / 100, rem = r % 100, ci = rem / 10, cj = rem % 10;
            float a  = gin[c][2 * ci][2 * cj],     bq = gin[c][2 * ci][2 * cj + 1];
            float d  = gin[c][2 * ci + 1][2 * cj], e  = gin[c][2 * ci + 1][2 * cj + 1];
            coef[c * 4 + 0][ci][cj] = 0.5f * (a + bq + d + e);   // LL
            coef[c * 4 + 1][ci][cj] = 0.5f * (a + bq - d - e);   // LH
            coef[c * 4 + 2][ci][cj] = 0.5f * (a - bq + d - e);   // HL
            coef[c * 4 + 3][ci][cj] = 0.5f * (a - bq - d + e);   // HH
        }
        __syncthreads();
        // ---- Phase C: 3x3 depthwise conv on 20 coeff channels, * wt_s
        for (int r = t; r < 1280; r += 256) {
            int ch = r >> 6, rem = r & 63, wi = rem >> 3, wj = rem & 7;
            const float* wt = g_wt_w + (size_t)(br * 20 + ch) * 9;
            float acc = 0.f;
#pragma unroll
            for (int u = 0; u < 3; ++u)
#pragma unroll
                for (int v = 0; v < 3; ++v)
                    acc += coef[ch][wi + u][wj + v] * wt[u * 3 + v];
            wconv[ch][wi][wj] = acc * g_wt_s[br * 20 + ch];
        }
        __syncthreads();
        // ---- Phase D: base dw3x3 + iDWT + 5x5 pointwise -> x20
        {
            int y = t >> 4, x = t & 15;
            int p = y & 1, q = x & 1, wi = y >> 1, wj = x >> 1;
            float sp  = p ? -0.5f : 0.5f;
            float sq  = q ? -0.5f : 0.5f;
            float spq = (p ^ q) ? -0.5f : 0.5f;
            float sum[5];
#pragma unroll
            for (int c = 0; c < 5; ++c) {
                const float* bw = g_base_w + (size_t)(br * 5 + c) * 9;
                float acc = 0.f;
#pragma unroll
                for (int u = 0; u < 3; ++u)
#pragma unroll
                    for (int v = 0; v < 3; ++v)
                        acc += gin[c][y + 1 + u][x + 1 + v] * bw[u * 3 + v];
                float base = (acc + g_base_b[br * 5 + c]) * g_base_s[br * 5 + c];
                float xt = 0.5f * wconv[c * 4 + 0][wi][wj]
                         + sp   * wconv[c * 4 + 1][wi][wj]
                         + sq   * wconv[c * 4 + 2][wi][wj]
                         + spq  * wconv[c * 4 + 3][wi][wj];
                sum[c] = base + xt;
            }
#pragma unroll
            for (int o = 0; o < 5; ++o) {
                float acc = 0.f;
#pragma unroll
                for (int c = 0; c < 5; ++c)
                    acc += g_pw_w[(br * 5 + o) * 5 + c] * sum[c];
                x20[br * 5 + o][t] = acc;
            }
        }
    }
    __syncthreads();
    // ---- Tail LN over 20 channels, in place
    {
        float v[20];
        float u = 0.f;
#pragma unroll
        for (int c = 0; c < 20; ++c) { v[c] = x20[c][t]; u += v[c]; }
        u *= (1.0f / 20.0f);
        float var = 0.f;
#pragma unroll
        for (int c = 0; c < 20; ++c) { float d = v[c] - u; var += d * d; }
        var *= (1.0f / 20.0f);
        float inv = rsqrtf(var + 1e-6f);
#pragma unroll
        for (int c = 0; c < 20; ++c)
            x20[c][t] = tail_ln_w[c] * (v[c] - u) * inv + tail_ln_b[c];
    }
    __syncthreads();
    // ---- Fused tail(20->8) + residual(9->8) GEMM via V_WMMA_F32_16X16X4_F32
    // M=16 (8 tail rows + 8 res rows), N=16 pixels (one tile row), K=32 in 8
    // chunks. B operand is branch-free LDS: x20 for k<20, xlc (incl. zero-pad
    // channels) for 20<=k<32 -> fully EXEC-uniform WMMA clause.
    {
        const int lane  = t & 31;
        const int wave  = t >> 5;
        const int nIdx  = lane & 15;            // M for A-fetch, N for B-fetch
        const int khalf = (lane >> 4) << 1;     // lane half selects K pair

        auto fetchB = [&](int k, int gy, int n) -> float {
            return (k < 20) ? x20[k][gy * 16 + n] : xlc[k - 20][gy * 16 + n];
        };

        for (int gsub = 0; gsub < 2; ++gsub) {
            const int gy = wave * 2 + gsub;     // pixel row within tile (0..15)
            v8f acc = {};
#pragma unroll
            for (int k0 = 0; k0 < 32; k0 += 4) {
                int kA = k0 + khalf;
                v2f a; a.x = wcomb[nIdx][kA]; a.y = wcomb[nIdx][kA + 1];
                v2f bb; bb.x = fetchB(kA, gy, nIdx); bb.y = fetchB(kA + 1, gy, nIdx);
                acc = __builtin_amdgcn_wmma_f32_16x16x4_f32(
                        false, a, false, bb, (short)0, acc, false, false);
            }
            // D layout: vgpr r, lanes 0-15 -> M=r (tail), lanes 16-31 -> M=r+8 (res)
#pragma unroll
            for (int r2 = 0; r2 < 8; ++r2) {
                float dv   = acc[r2];
                float resv = __shfl(dv, nIdx + 16, 32);   // pull residual half
                if (lane < 16) {
                    float tval = dv + tail_b[r2];
                    float oval = gelu_exact(tval) + resv + res_b[r2];
                    out[(((size_t)b * 8 + r2) * HL + (oy + gy)) * HL + ox + lane] = oval;
                }
            }
        }
    }
}

// ---------------------------------------------------------------------------
extern "C" void kernel_launch(void* const* d_in, const int* in_sizes, int n_in,
                              void* d_out, int out_size, void* d_ws, size_t ws_size,
                              hipStream_t stream) {
    (void)in_sizes; (void)n_in; (void)out_size; (void)ws_size;
    const float* xh        = (const float*)d_in[0];
    const float* xl        = (const float*)d_in[1];
    const float* mask      = (const float*)d_in[2];
    const float* pre_w     = (const float*)d_in[3];
    const float* pre_b     = (const float*)d_in[4];
    const float* g_ln_w    = (const float*)d_in[5];
    const float* g_ln_b    = (const float*)d_in[6];
    const float* g_base_w  = (const float*)d_in[7];
    const float* g_base_b  = (const float*)d_in[8];
    const float* g_base_s  = (const float*)d_in[9];
    const float* g_wt_w    = (const float*)d_in[10];
    const float* g_wt_s    = (const float*)d_in[11];
    const float* g_pw_w    = (const float*)d_in[12];
    const float* tail_ln_w = (const float*)d_in[13];
    const float* tail_ln_b = (const float*)d_in[14];
    const float* tail_w    = (const float*)d_in[15];
    const float* tail_b    = (const float*)d_in[16];
    const float* res_w     = (const float*)d_in[17];
    const float* res_b     = (const float*)d_in[18];
    float* out = (float*)d_out;

    float* xh_p = (float*)d_ws;   // [8, 8, 256, 256] = 16 MB scratch

    // Kernel 0: pre-projection
    int total0 = NB * 8 * HHS * HHS;              // 4,194,304
    pre_project_kernel<<<total0 / 256, 256, 0, stream>>>(xh, pre_w, pre_b, xh_p);

    // Fused kernel: one block per 16x16 tile per batch
    dim3 grid(HL / 16, HL / 16, NB);
    fused_kernel<<<grid, 256, 0, stream>>>(
        xh_p, xl, mask, g_ln_w, g_ln_b, g_base_w, g_base_b, g_base_s,
        g_wt_w, g_wt_s, g_pw_w, tail_ln_w, tail_ln_b, tail_w, tail_b,
        res_w, res_b, out);
}